// GramSchmidt03_19980187861911
// MI455X (gfx1250) — compile-verified
//
#include <hip/hip_runtime.h>

typedef __attribute__((ext_vector_type(2))) float v2f;
typedef __attribute__((ext_vector_type(4))) float v4f;
typedef __attribute__((ext_vector_type(8))) float v8f;
typedef __attribute__((ext_vector_type(4))) int   v4i;

#define KD 32
#define LD 256
#define XS_STRIDE 260   // pad rows: 4 banks apart -> 16 parallel rows hit 16 distinct banks
#define GS_STRIDE 33
#define TP_STRIDE 34    // even stride keeps float2 loads 8B-aligned

#if defined(__AMDGCN__) && __has_builtin(__builtin_amdgcn_global_load_async_to_lds_b128)
#define USE_ASYNC_LDS 1
#else
#define USE_ASYNC_LDS 0
#endif

// One workgroup (128 threads = 4 waves) per batch.
// Phase 0: X (32x256 fp32) -> LDS via GLOBAL_LOAD_ASYNC_TO_LDS_B128 (ASYNCcnt path)
// Phase 1: G = X * X^T via v_wmma_f32_16x16x4_f32 (4 tiles, one per wave)
// Phase 2: Gram-domain classical GS recursion (wave 0):
//          T unit-lower-triangular with Q = T*X, W = T*G, coef c_ik = W[k][i]/n[k],
//          n[k] = sum_j T[k][j]*W[k][j]  (== q_k . q_k exactly in the Gram domain)
// Phase 3: T' = diag(1/sqrt(n)) * T
// Phase 4: Y = T' * X via WMMA, store to global
__global__ __launch_bounds__(128) void gram_schmidt_wmma(const float* __restrict__ x,
                                                         float* __restrict__ out) {
  __shared__ __align__(16) float Xs[KD * XS_STRIDE];
  __shared__ float Gs[KD * GS_STRIDE];
  __shared__ float Ts[KD * GS_STRIDE];
  __shared__ float Ws[KD * GS_STRIDE];
  __shared__ float Tp[KD * TP_STRIDE];
  __shared__ float ns[KD];

  const int b    = blockIdx.x;
  const int tid  = threadIdx.x;
  const int lane = tid & 31;
  const int wave = tid >> 5;
  const float* xb = x + (size_t)b * (KD * LD);
  float* ob       = out + (size_t)b * (KD * LD);

  // ---- Phase 0: X -> LDS (padded rows). Async memory->LDS path when available.
#if USE_ASYNC_LDS
  {
#pragma unroll
    for (int it = 0; it < (KD * LD / 4) / 128; ++it) {
      int f   = it * 128 + tid;
      int row = f >> 6;      // 64 float4 per 256-float row
      int c4  = f & 63;
      v4i* gp = (v4i*)(xb + (size_t)f * 4);
      v4i* lp = (v4i*)(&Xs[row * XS_STRIDE + c4 * 4]);
      __builtin_amdgcn_global_load_async_to_lds_b128(gp, lp, 0, 0);
    }
#if __has_builtin(__builtin_amdgcn_s_wait_asynccnt)
    __builtin_amdgcn_s_wait_asynccnt(0);
#else
    asm volatile("s_wait_asynccnt 0" ::: "memory");
#endif
  }
#else
  for (int f = tid; f < (KD * LD / 4); f += 128) {
    int row = f >> 6;
    int c4  = f & 63;
    v4f v = *(const v4f*)(xb + (size_t)f * 4);
    *(v4f*)(&Xs[row * XS_STRIDE + c4 * 4]) = v;
  }
#endif
  __syncthreads();

  // Per-lane WMMA operand geometry (16x16x4 f32):
  // A: lanes 0-15 hold M=lane, K=k0..k0+1 with k0=0; lanes 16-31 same M, k0=2.
  // B mirrors with N=lane&15. C/D: VGPR r -> M = r + 8*(lane>=16), N = lane&15.
  const int half = lane >> 4;        // 0 or 1
  const int k0   = half << 1;        // 0 or 2
  const int mn   = lane & 15;

  // ---- Phase 1: G = X * X^T, one 16x16 tile per wave ----
  {
    const int mt = (wave >> 1) & 1;
    const int nt = wave & 1;
    const int rA = mt * 16 + mn;
    const int rB = nt * 16 + mn;
    v8f acc = {};
#pragma unroll 4
    for (int l0 = 0; l0 < LD; l0 += 4) {
      v2f a  = *(const v2f*)(&Xs[rA * XS_STRIDE + l0 + k0]);
      v2f bm = *(const v2f*)(&Xs[rB * XS_STRIDE + l0 + k0]);
      acc = __builtin_amdgcn_wmma_f32_16x16x4_f32(false, a, false, bm,
                                                  (short)0, acc, false, false);
    }
    const int mbase = mt * 16 + half * 8;
    const int cn    = nt * 16 + mn;
#pragma unroll
    for (int r = 0; r < 8; ++r)
      Gs[(mbase + r) * GS_STRIDE + cn] = acc[r];
  }
  __syncthreads();

  // ---- Phase 2: Gram-domain recursion (wave 0; lane j owns column j) ----
  if (wave == 0) {
    int j = lane;
    Ts[j] = (j == 0) ? 1.0f : 0.0f;
    float w0 = Gs[j];
    Ws[j] = w0;
    if (j == 0) ns[0] = w0;        // T0 = e0  ->  n0 = W[0][0]
  }
  __syncthreads();

  for (int i = 1; i < KD; ++i) {
    if (wave == 0) {
      int j = lane;
      float tij = (j == i) ? 1.0f : 0.0f;
      float wij = Gs[i * GS_STRIDE + j];
      for (int k = 0; k < i; ++k) {
        float c = Ws[k * GS_STRIDE + i] / ns[k];   // broadcast LDS reads
        tij -= c * Ts[k * GS_STRIDE + j];
        wij -= c * Ws[k * GS_STRIDE + j];
      }
      Ts[i * GS_STRIDE + j] = tij;
      Ws[i * GS_STRIDE + j] = wij;
      float p = tij * wij;                         // n_i = T_i . W_i
#pragma unroll
      for (int off = 16; off > 0; off >>= 1) p += __shfl_xor(p, off, 32);
      if (j == 0) ns[i] = p;
    }
    __syncthreads();
  }

  // ---- Phase 3: fold inv-norm into T ----
  for (int e = tid; e < KD * KD; e += 128) {
    int i = e >> 5, j = e & 31;
    Tp[i * TP_STRIDE + j] = Ts[i * GS_STRIDE + j] * (1.0f / __builtin_sqrtf(ns[i]));
  }
  __syncthreads();

  // ---- Phase 4: Y = T' * X, 32 output tiles round-robined over 4 waves ----
  for (int t = wave; t < 32; t += 4) {
    const int mt = t & 1;        // row block (0/1)
    const int lt = t >> 1;       // 16-wide column block (0..15)
    const int rA = mt * 16 + mn;
    const int nB = lt * 16 + mn;
    v8f acc = {};
#pragma unroll
    for (int j0 = 0; j0 < KD; j0 += 4) {
      v2f a = *(const v2f*)(&Tp[rA * TP_STRIDE + j0 + k0]);
      v2f bm;
      bm.x = Xs[(j0 + k0) * XS_STRIDE + nB];
      bm.y = Xs[(j0 + k0 + 1) * XS_STRIDE + nB];
      acc = __builtin_amdgcn_wmma_f32_16x16x4_f32(false, a, false, bm,
                                                  (short)0, acc, false, false);
    }
    const int mbase = mt * 16 + half * 8;
    const int cn    = lt * 16 + mn;
#pragma unroll
    for (int r = 0; r < 8; ++r)
      ob[(size_t)(mbase + r) * LD + cn] = acc[r];
  }
}

extern "C" void kernel_launch(void* const* d_in, const int* in_sizes, int n_in,
                              void* d_out, int out_size, void* d_ws, size_t ws_size,
                              hipStream_t stream) {
  (void)n_in; (void)out_size; (void)d_ws; (void)ws_size;
  const float* x = (const float*)d_in[0];
  float* out = (float*)d_out;
  int nbatch = in_sizes[0] / (KD * LD);
  gram_schmidt_wmma<<<nbatch, 128, 0, stream>>>(x, out);
}